// MuonLonghornMixer_61641370632616
// MI455X (gfx1250) — compile-verified
//
#include <hip/hip_runtime.h>
#include <hip/hip_bf16.h>

// MuonLonghornMixer for MI455X (gfx1250), fp32 end-to-end using
// V_WMMA_F32_16X16X4_F32 for all matrix work (big GEMMs + per-step
// Newton-Schulz Gram matrices in the scan).
//
// Shapes: B=2, L=2048, DM=1024, DI=2048, N=16, R=64, W=4.

#define B_  2
#define L_  2048
#define DM_ 1024
#define DI_ 2048
#define NS_ 16
#define R_  64
#define W_  4

#define BETA_  0.9f
#define ALPHA_ 1.0f
#define NSA_   3.4445f
#define NSB_   (-4.7750f)
#define NSC_   2.0315f

typedef __attribute__((ext_vector_type(2))) float v2f;
typedef __attribute__((ext_vector_type(8))) float v8f;

__device__ __forceinline__ float silu_f(float x) {
    return x / (1.0f + __expf(-x));
}

// ---------------------------------------------------------------------------
// Generic fp32 WMMA GEMM:  C[M,N] = A[M,K(lda)] * W[N,K]^T  (all row-major).
// Each wave computes a 16 x (16*NT) strip: one A fragment feeds NT
// independent WMMA accumulator chains (pipelined, A traffic amortized).
// EPI==1: C = softplus(C + bias[n])  (dt head)
// ---------------------------------------------------------------------------
template <int EPI, int NT>
__global__ void __launch_bounds__(256)
gemm_f32_wmma(const float* __restrict__ A, const float* __restrict__ W,
              const float* __restrict__ bias, float* __restrict__ C,
              int M, int N, int K, int lda) {
    const int wavesPerBlock = blockDim.x >> 5;
    const int wave   = blockIdx.x * wavesPerBlock + (threadIdx.x >> 5);
    const int lane   = threadIdx.x & 31;
    const int half   = lane >> 4;   // 0: lanes 0-15, 1: lanes 16-31
    const int l16    = lane & 15;
    const int nstrip = N / (16 * NT);
    const int tileM  = (wave / nstrip) << 4;
    const int tileN  = (wave % nstrip) * (16 * NT);
    if (tileM >= M) return;         // uniform per-wave

    const float* Arow = A + (size_t)(tileM + l16) * lda;
    const float* Wrow = W + (size_t)(tileN + l16) * K;

    v8f acc[NT];
#pragma unroll
    for (int t = 0; t < NT; ++t) acc[t] = (v8f){};

    for (int k0 = 0; k0 < K; k0 += 4) {
        const int ka = k0 + 2 * half;             // lanes 16-31 take K=2,3
        float2 av = *(const float2*)(Arow + ka);  // A[M=l16][ka, ka+1]
        v2f a = {av.x, av.y};
#pragma unroll
        for (int t = 0; t < NT; ++t) {
            float2 bv = *(const float2*)(Wrow + (size_t)t * 16 * K + ka);
            v2f b = {bv.x, bv.y};                 // B[K][N=l16] = W[N][K]
            acc[t] = __builtin_amdgcn_wmma_f32_16x16x4_f32(
                false, a, false, b, (short)0, acc[t], false, false);
        }
    }

#pragma unroll
    for (int t = 0; t < NT; ++t) {
        const int n = tileN + t * 16 + l16;
#pragma unroll
        for (int r = 0; r < 8; ++r) {
            const int m = tileM + r + 8 * half;   // C/D layout: M = r + 8*half
            float v = acc[t][r];
            if (EPI == 1) {                       // + bias, softplus
                v += bias[n];
                v = (v > 20.0f) ? v : log1pf(__expf(v));
            }
            C[(size_t)m * N + n] = v;
        }
    }
}

// ---------------------------------------------------------------------------
// Depthwise causal conv1d (W=4) + SiLU.  xz: (B*L, 2*DI), x part = cols [0,DI)
// ---------------------------------------------------------------------------
__global__ void __launch_bounds__(256)
conv_silu_kernel(const float* __restrict__ xz, const float* __restrict__ cw,
                 const float* __restrict__ cb, float* __restrict__ xc) {
    const size_t idx = (size_t)blockIdx.x * blockDim.x + threadIdx.x;
    if (idx >= (size_t)B_ * L_ * DI_) return;
    const int d  = (int)(idx % DI_);
    const int bl = (int)(idx / DI_);
    const int l  = bl % L_;
    const int b  = bl / L_;
    float acc = cb[d];
#pragma unroll
    for (int w = 0; w < W_; ++w) {
        const int li = l - (W_ - 1) + w;
        if (li >= 0)
            acc += cw[d * W_ + w] *
                   xz[((size_t)(b * L_ + li)) * (2 * DI_) + d];
    }
    xc[idx] = silu_f(acc);
}

// ---------------------------------------------------------------------------
// Longhorn scan with Muon (Newton-Schulz) update.  One workgroup per batch.
// 1024 threads; thread t owns rows d = 2t, 2t+1.
//   S and momentum m live in LDS (~258 KB of the WGP's 320 KB); k/q stay in
//   LDS too (broadcast reads), so per-thread register pressure stays small
//   and the 2048-step sequential critical path has no scratch spills.
// Per step the 16x16 Gram G = m^T m is computed with WMMA f32 16x16x4 over
// each wave's 64-row slice (two independent accumulator chains), reduced via
// ds_add_f32 atomics; the 1/||m|| normalization is folded in as inv^2 when
// forming Bm = b*A + c*A@A with A = inv^2 * G (Bm is symmetric).
// ---------------------------------------------------------------------------
__global__ void __launch_bounds__(1024)
scan_kernel(const float* __restrict__ xc,    // (B*L, DI) conv+silu output
            const float* __restrict__ xdbl,  // (B*L, 96): [64,80)=k [80,96)=q
            const float* __restrict__ dt,    // (B*L, DI) softplus(dt)
            float* __restrict__ yout) {      // (B*L, DI)
    extern __shared__ float smem[];
    float* Sbuf  = smem;                  // DI*16 floats: the state S
    float* stage = Sbuf + DI_ * NS_;      // DI*16 floats: the momentum m
    float* Aacc  = stage + DI_ * NS_;     // 256  (Gram accumulator)
    float* Bmat  = Aacc + 256;            // 256  (symmetric b*A + c*A@A)
    float* nrm   = Bmat + 256;            // 8    ([0] = ||m||_F^2)
    float* kq    = nrm + 8;               // 32   (k then q)

    const int tid   = threadIdx.x;
    const int wave  = tid >> 5;
    const int lane  = tid & 31;
    const int half  = lane >> 4;
    const int l16   = lane & 15;
    const int b     = blockIdx.x;
    const int d0    = tid * 2;            // first owned row
    const int dbase = wave * 64;          // wave's Gram K-slice

    // zero S and m (this workgroup owns them)
#pragma unroll
    for (int r = 0; r < 2; ++r)
#pragma unroll
        for (int n = 0; n < NS_; ++n) {
            Sbuf [(d0 + r) * NS_ + n] = 0.0f;
            stage[(d0 + r) * NS_ + n] = 0.0f;
        }

    for (int l = 0; l < L_; ++l) {
        const size_t bl = (size_t)b * L_ + l;

        __syncthreads();                          // prior step fully consumed
        if (tid < 256) Aacc[tid] = 0.0f;
        if (tid == 0)  nrm[0] = 0.0f;
        if (tid < 32)  kq[tid] = xdbl[bl * (R_ + 2 * NS_) + R_ + tid];
        __syncthreads();

        // ||k||^2 from LDS broadcast reads (k is workgroup-uniform)
        float ksq = 0.0f;
#pragma unroll
        for (int n = 0; n < NS_; ++n) ksq += kq[n] * kq[n];

        // momentum update (LDS-resident S, m, k) + ||m||^2 partial
        float sumsq = 0.0f;
#pragma unroll
        for (int r = 0; r < 2; ++r) {
            const int d = d0 + r;
            const float xv  = xc[bl * DI_ + d];
            const float dtv = dt[bl * DI_ + d];
            const float eps = dtv / (1.0f + dtv * ksq);
            float err = -xv;
#pragma unroll
            for (int n = 0; n < NS_; ++n) err += Sbuf[d * NS_ + n] * kq[n];
            const float ce = eps * err;
#pragma unroll
            for (int n = 0; n < NS_; ++n) {
                const float mv = BETA_ * stage[d * NS_ + n] + ce * kq[n];
                stage[d * NS_ + n] = mv;
                sumsq += mv * mv;
            }
        }
        // wave32 reduction, one LDS atomic per wave
#pragma unroll
        for (int off = 16; off > 0; off >>= 1)
            sumsq += __shfl_xor(sumsq, off);
        if (lane == 0) atomicAdd(&nrm[0], sumsq);
        __syncthreads();                          // m + ||m||^2 visible

        // Gram: G[i][j] = sum_d m[d][i]*m[d][j] over this wave's 64 rows.
        // Two independent WMMA chains so ds_loads overlap WMMA execution.
        v8f p0 = (v8f){}, p1 = (v8f){};
        for (int kc = 0; kc < 64; kc += 8) {
            const int dk0 = dbase + kc + 2 * half;
            const int dk1 = dbase + kc + 4 + 2 * half;
            v2f a0 = { stage[dk0 * NS_ + l16], stage[(dk0 + 1) * NS_ + l16] };
            v2f a1 = { stage[dk1 * NS_ + l16], stage[(dk1 + 1) * NS_ + l16] };
            p0 = __builtin_amdgcn_wmma_f32_16x16x4_f32(
                false, a0, false, a0, (short)0, p0, false, false);
            p1 = __builtin_amdgcn_wmma_f32_16x16x4_f32(
                false, a1, false, a1, (short)0, p1, false, false);
        }
#pragma unroll
        for (int r = 0; r < 8; ++r)
            atomicAdd(&Aacc[(r + 8 * half) * NS_ + l16], p0[r] + p1[r]);
        __syncthreads();

        // Bm = b*A + c*(A@A), A = inv^2 * G   (threads 0..255)
        const float inv  = 1.0f / (sqrtf(nrm[0]) + 1e-7f);
        const float inv2 = inv * inv;
        if (tid < 256) {
            const int i = tid >> 4, j = tid & 15;
            float s = 0.0f;
#pragma unroll
            for (int t = 0; t < NS_; ++t)
                s += Aacc[i * NS_ + t] * Aacc[t * NS_ + j];
            Bmat[tid] = NSB_ * (inv2 * Aacc[tid]) + NSC_ * (inv2 * inv2 * s);
        }
        __syncthreads();

        // S -= alpha * (a*m_hat + m_hat @ Bm); y = S @ q   (Bm symmetric)
#pragma unroll
        for (int r = 0; r < 2; ++r) {
            const int d = d0 + r;
            float mh[NS_];
#pragma unroll
            for (int n = 0; n < NS_; ++n) mh[n] = stage[d * NS_ + n] * inv;
            float y = 0.0f;
#pragma unroll
            for (int n = 0; n < NS_; ++n) {
                float ns = NSA_ * mh[n];
#pragma unroll
                for (int j = 0; j < NS_; ++j)
                    ns += Bmat[n * NS_ + j] * mh[j];   // contiguous in j
                const float sv = Sbuf[d * NS_ + n] - ALPHA_ * ns;
                Sbuf[d * NS_ + n] = sv;
                y += sv * kq[NS_ + n];                 // q from LDS
            }
            yout[bl * DI_ + d] = y;
        }
    }
}

// ---------------------------------------------------------------------------
// g = (y + D*xc) * silu(z)
// ---------------------------------------------------------------------------
__global__ void __launch_bounds__(256)
gate_kernel(const float* __restrict__ yb, const float* __restrict__ xc,
            const float* __restrict__ xz, const float* __restrict__ Dv,
            float* __restrict__ g) {
    const size_t idx = (size_t)blockIdx.x * blockDim.x + threadIdx.x;
    if (idx >= (size_t)B_ * L_ * DI_) return;
    const int d  = (int)(idx % DI_);
    const size_t bl = idx / DI_;
    const float z = xz[bl * (2 * DI_) + DI_ + d];
    g[idx] = (yb[idx] + Dv[d] * xc[idx]) * silu_f(z);
}

// ---------------------------------------------------------------------------
extern "C" void kernel_launch(void* const* d_in, const int* in_sizes, int n_in,
                              void* d_out, int out_size, void* d_ws, size_t ws_size,
                              hipStream_t stream) {
    (void)in_sizes; (void)n_in; (void)out_size; (void)ws_size;
    const float* hidden     = (const float*)d_in[0];
    const float* in_proj_w  = (const float*)d_in[1];
    const float* conv_w     = (const float*)d_in[2];
    const float* conv_b     = (const float*)d_in[3];
    const float* x_proj_w   = (const float*)d_in[4];
    const float* dt_head_w  = (const float*)d_in[5];
    const float* dt_head_b  = (const float*)d_in[6];
    const float* out_proj_w = (const float*)d_in[7];
    const float* Dvec       = (const float*)d_in[8];
    float* out = (float*)d_out;

    const int BL = B_ * L_;
    float* ws   = (float*)d_ws;
    float* xz   = ws;                                     // (BL, 2*DI)
    float* xc   = xz   + (size_t)BL * (2 * DI_);          // (BL, DI)
    float* xdbl = xc   + (size_t)BL * DI_;                // (BL, 96)
    float* dtb  = xdbl + (size_t)BL * (R_ + 2 * NS_);     // (BL, DI)
    float* yb   = dtb  + (size_t)BL * DI_;                // (BL, DI)
    float* gb   = yb   + (size_t)BL * DI_;                // (BL, DI)

    // 1) in_proj: xz = hidden @ in_proj_w^T   (4096 x 4096, K=1024)
    {
        int strips = (BL / 16) * ((2 * DI_) / 64);
        gemm_f32_wmma<0, 4><<<dim3(strips / 8), 256, 0, stream>>>(
            hidden, in_proj_w, nullptr, xz, BL, 2 * DI_, DM_, DM_);
    }
    // 2) depthwise causal conv + SiLU
    conv_silu_kernel<<<dim3((BL * DI_) / 256), 256, 0, stream>>>(
        xz, conv_w, conv_b, xc);
    // 3a) x_dbl = xc @ x_proj_w^T   (4096 x 96, K=2048)
    {
        int strips = (BL / 16) * ((R_ + 2 * NS_) / 32);
        gemm_f32_wmma<0, 2><<<dim3(strips / 8), 256, 0, stream>>>(
            xc, x_proj_w, nullptr, xdbl, BL, R_ + 2 * NS_, DI_, DI_);
    }
    // 3b) dt = softplus(dt_low @ dt_head_w^T + b)  (4096 x 2048, K=64, lda=96)
    {
        int strips = (BL / 16) * (DI_ / 64);
        gemm_f32_wmma<1, 4><<<dim3(strips / 8), 256, 0, stream>>>(
            xdbl, dt_head_w, dt_head_b, dtb, BL, DI_, R_, R_ + 2 * NS_);
    }
    // 4) sequential Longhorn/Muon scan: one WGP-resident workgroup per batch
    //    LDS: S (128KB) + m (128KB) + Gram/Bm/norm/kq (~2KB) < 320KB WGP LDS
    {
        size_t smem = ((size_t)2 * DI_ * NS_ + 256 + 256 + 8 + 32) * sizeof(float);
        hipFuncSetAttribute(reinterpret_cast<const void*>(scan_kernel),
                            hipFuncAttributeMaxDynamicSharedMemorySize,
                            (int)smem);
        scan_kernel<<<dim3(B_), 1024, smem, stream>>>(xc, xdbl, dtb, yb);
    }
    // 5) gating
    gate_kernel<<<dim3((BL * DI_) / 256), 256, 0, stream>>>(
        yb, xc, xz, Dvec, gb);
    // 6) out_proj: out = g @ out_proj_w^T   (4096 x 1024, K=2048)
    {
        int strips = (BL / 16) * (DM_ / 64);
        gemm_f32_wmma<0, 4><<<dim3(strips / 8), 256, 0, stream>>>(
            gb, out_proj_w, nullptr, out, BL, DM_, DI_, DI_);
    }
}